// GCORN_6803228197263
// MI455X (gfx1250) — compile-verified
//
#include <hip/hip_runtime.h>
#include <hip/hip_bf16.h>
#include <math.h>

#define NNODES 12000
#define NEDGES 384000
#define HID    128
#define NOUT   40
#define BJ_ITERS 25

typedef __attribute__((ext_vector_type(2))) float v2f;
typedef __attribute__((ext_vector_type(8))) float v8f;

// ---------------------------------------------------------------------------
// degree / normalization
// ---------------------------------------------------------------------------
__global__ void k_degree(const int* __restrict__ er, float* __restrict__ deg, int E) {
    int e = blockIdx.x * blockDim.x + threadIdx.x;
    if (e < E) atomicAdd(&deg[er[e]], 1.0f);
}

__global__ void k_dinv(const float* __restrict__ deg, float* __restrict__ dinv, int n) {
    int i = blockIdx.x * blockDim.x + threadIdx.x;
    if (i < n) {
        float d = deg[i];
        dinv[i] = (d > 0.0f) ? (1.0f / sqrtf(d)) : 0.0f;
    }
}

// ---------------------------------------------------------------------------
// Bjorck init: V = W^T / sqrt(max_rowsum(|W|) * max_colsum(|W|))
// single block, 128 threads
// ---------------------------------------------------------------------------
__global__ void k_init_bjorck(const float* __restrict__ W, float* __restrict__ V) {
    __shared__ float rs[HID];
    __shared__ float cs[HID];
    __shared__ float sc;
    int t = threadIdx.x;
    float r = 0.0f, c = 0.0f;
    for (int j = 0; j < HID; ++j) {
        r += fabsf(W[t * HID + j]);
        c += fabsf(W[j * HID + t]);
    }
    rs[t] = r; cs[t] = c;
    __syncthreads();
    if (t == 0) {
        float rm = 0.0f, cm = 0.0f;
        for (int i = 0; i < HID; ++i) {
            rm = fmaxf(rm, rs[i]);
            cm = fmaxf(cm, cs[i]);
        }
        sc = sqrtf(rm * cm);
    }
    __syncthreads();
    float inv = 1.0f / sc;
    for (int o = 0; o < HID; ++o)
        V[t * HID + o] = W[o * HID + t] * inv;   // transpose + scale
}

// ---------------------------------------------------------------------------
// WMMA f32 16x16x4 helpers. One wave computes one 16x16 output tile.
// A-frag: lane holds row (lane&15), k-pair khalf = (lane>>4)*2
// B-frag: same k striping, column = lane&15
// C/D:   VGPR v -> rows v / v+8, lane -> column
// ---------------------------------------------------------------------------

// C[M,128] = A[M,128] @ B[128,128]   (row-major, M multiple of 16)
__global__ void k_gemm_k128n128(const float* __restrict__ A,
                                const float* __restrict__ B,
                                float* __restrict__ C, int M) {
    int wid  = (blockIdx.x * blockDim.x + threadIdx.x) >> 5;
    int lane = threadIdx.x & 31;
    int tiles_m = M >> 4;
    int tm = wid >> 3;          // 8 tiles along N (128/16)
    int tn = wid & 7;
    if (tm >= tiles_m) return;  // wave-uniform

    int m     = (lane & 15);
    int khalf = (lane >> 4) * 2;
    int row   = tm * 16 + m;
    int col   = tn * 16 + m;

    v8f acc = {};
    for (int k = 0; k < 128; k += 4) {
        v2f a, b;
        a.x = A[(size_t)row * 128 + k + khalf];
        a.y = A[(size_t)row * 128 + k + khalf + 1];
        b.x = B[(size_t)(k + khalf) * 128 + col];
        b.y = B[(size_t)(k + khalf + 1) * 128 + col];
        acc = __builtin_amdgcn_wmma_f32_16x16x4_f32(false, a, false, b,
                                                    (short)0, acc, false, false);
    }
    int base_m = tm * 16 + ((lane >= 16) ? 8 : 0);
    int n = tn * 16 + m;
#pragma unroll
    for (int v = 0; v < 8; ++v)
        C[(size_t)(base_m + v) * 128 + n] = acc[v];
}

// S = V^T @ V   (V, S are 128x128)
__global__ void k_ata128(const float* __restrict__ V, float* __restrict__ S) {
    int wid  = (blockIdx.x * blockDim.x + threadIdx.x) >> 5;
    int lane = threadIdx.x & 31;
    int ti = wid >> 3, tj = wid & 7;
    if (ti >= 8) return;

    int m     = lane & 15;
    int khalf = (lane >> 4) * 2;
    v8f acc = {};
    for (int k = 0; k < 128; k += 4) {
        v2f a, b;
        // A = V^T : element (i, kk) = V[kk][i]
        a.x = V[(size_t)(k + khalf)     * 128 + ti * 16 + m];
        a.y = V[(size_t)(k + khalf + 1) * 128 + ti * 16 + m];
        b.x = V[(size_t)(k + khalf)     * 128 + tj * 16 + m];
        b.y = V[(size_t)(k + khalf + 1) * 128 + tj * 16 + m];
        acc = __builtin_amdgcn_wmma_f32_16x16x4_f32(false, a, false, b,
                                                    (short)0, acc, false, false);
    }
    int base_m = ti * 16 + ((lane >= 16) ? 8 : 0);
    int n = tj * 16 + m;
#pragma unroll
    for (int v = 0; v < 8; ++v)
        S[(size_t)(base_m + v) * 128 + n] = acc[v];
}

// Mo = 0.375*(S@S) - 1.25*S + 1.875*I   (128x128)
__global__ void k_poly128(const float* __restrict__ S, float* __restrict__ Mo) {
    int wid  = (blockIdx.x * blockDim.x + threadIdx.x) >> 5;
    int lane = threadIdx.x & 31;
    int ti = wid >> 3, tj = wid & 7;
    if (ti >= 8) return;

    int m     = lane & 15;
    int khalf = (lane >> 4) * 2;
    int row   = ti * 16 + m;
    int col   = tj * 16 + m;
    v8f acc = {};
    for (int k = 0; k < 128; k += 4) {
        v2f a, b;
        a.x = S[(size_t)row * 128 + k + khalf];
        a.y = S[(size_t)row * 128 + k + khalf + 1];
        b.x = S[(size_t)(k + khalf)     * 128 + col];
        b.y = S[(size_t)(k + khalf + 1) * 128 + col];
        acc = __builtin_amdgcn_wmma_f32_16x16x4_f32(false, a, false, b,
                                                    (short)0, acc, false, false);
    }
    int base_m = ti * 16 + ((lane >= 16) ? 8 : 0);
    int n = tj * 16 + m;
#pragma unroll
    for (int v = 0; v < 8; ++v) {
        int r = base_m + v;
        float val = 0.375f * acc[v] - 1.25f * S[(size_t)r * 128 + n];
        if (r == n) val += 1.875f;
        Mo[(size_t)r * 128 + n] = val;
    }
}

// ---------------------------------------------------------------------------
// SpMM scatter: Hout[c] += dinv[c]*dinv[r]*Hin[r]   (one wave per edge)
// ---------------------------------------------------------------------------
__global__ void k_spmm(const int* __restrict__ er, const int* __restrict__ ec,
                       const float* __restrict__ dinv,
                       const float* __restrict__ Hin, float* __restrict__ Hout,
                       int E) {
    int wid  = (blockIdx.x * blockDim.x + threadIdx.x) >> 5;
    int lane = threadIdx.x & 31;
    if (wid >= E) return;
    int r = er[wid], c = ec[wid];
    float w = dinv[r] * dinv[c];
    const float4* src = (const float4*)(Hin + (size_t)r * HID);
    float4 v = src[lane];
    float* dst = Hout + (size_t)c * HID + lane * 4;
    atomicAdd(dst + 0, w * v.x);
    atomicAdd(dst + 1, w * v.y);
    atomicAdd(dst + 2, w * v.z);
    atomicAdd(dst + 3, w * v.w);
}

__global__ void k_relu(float* __restrict__ p, int n) {
    int i = blockIdx.x * blockDim.x + threadIdx.x;
    if (i < n) p[i] = fmaxf(p[i], 0.0f);
}

// ---------------------------------------------------------------------------
// Head: logits = H @ lin_w^T + b, then log_softmax.  One block (64 thr) per node
// ---------------------------------------------------------------------------
__global__ void k_head(const float* __restrict__ H, const float* __restrict__ LW,
                       const float* __restrict__ LB, float* __restrict__ out) {
    __shared__ float h[HID];
    __shared__ float z[NOUT];
    __shared__ float mred, sred;
    int n = blockIdx.x;
    int t = threadIdx.x;
    h[t]      = H[(size_t)n * HID + t];
    h[t + 64] = H[(size_t)n * HID + t + 64];
    __syncthreads();
    if (t < NOUT) {
        float acc = LB[t];
        const float* w = LW + (size_t)t * HID;
        for (int k = 0; k < HID; ++k) acc += h[k] * w[k];
        z[t] = acc;
    }
    __syncthreads();
    if (t == 0) {
        float m = -3.402823466e38f;
        for (int o = 0; o < NOUT; ++o) m = fmaxf(m, z[o]);
        float s = 0.0f;
        for (int o = 0; o < NOUT; ++o) s += expf(z[o] - m);
        mred = m; sred = logf(s);
    }
    __syncthreads();
    if (t < NOUT) out[(size_t)n * NOUT + t] = z[t] - mred - sred;
}

// ---------------------------------------------------------------------------
extern "C" void kernel_launch(void* const* d_in, const int* in_sizes, int n_in,
                              void* d_out, int out_size, void* d_ws, size_t ws_size,
                              hipStream_t stream) {
    const float* x   = (const float*)d_in[0];
    const float* W1  = (const float*)d_in[1];
    const float* W2  = (const float*)d_in[2];
    const float* lw  = (const float*)d_in[3];
    const float* lb  = (const float*)d_in[4];
    const int*   ei  = (const int*)d_in[5];
    const int*   er  = ei;            // edge_index[0]
    const int*   ec  = ei + NEDGES;   // edge_index[1]
    float* out = (float*)d_out;

    float* ws = (float*)d_ws;
    size_t o = 0;
    float* deg  = ws + o; o += 12032;
    float* dinv = ws + o; o += 12032;
    float* V1a  = ws + o; o += HID * HID;
    float* V1b  = ws + o; o += HID * HID;
    float* V2a  = ws + o; o += HID * HID;
    float* V2b  = ws + o; o += HID * HID;
    float* S    = ws + o; o += HID * HID;
    float* Mm   = ws + o; o += HID * HID;
    float* XW   = ws + o; o += (size_t)NNODES * HID;
    float* AX   = ws + o; o += (size_t)NNODES * HID;
    float* HW   = ws + o; o += (size_t)NNODES * HID;
    float* AH   = ws + o; o += (size_t)NNODES * HID;

    // --- degree & D^{-1/2} ---
    hipMemsetAsync(deg, 0, NNODES * sizeof(float), stream);
    k_degree<<<(NEDGES + 255) / 256, 256, 0, stream>>>(er, deg, NEDGES);
    k_dinv<<<(NNODES + 255) / 256, 256, 0, stream>>>(deg, dinv, NNODES);

    // --- Bjorck orthonormalization: B = bjorck(W^T / s), 25 iters ---
    float* B1;
    {
        k_init_bjorck<<<1, 128, 0, stream>>>(W1, V1a);
        float* cur = V1a; float* nxt = V1b;
        for (int it = 0; it < BJ_ITERS; ++it) {
            k_ata128 <<<8, 256, 0, stream>>>(cur, S);
            k_poly128<<<8, 256, 0, stream>>>(S, Mm);
            k_gemm_k128n128<<<8, 256, 0, stream>>>(cur, Mm, nxt, HID);
            float* t = cur; cur = nxt; nxt = t;
        }
        B1 = cur;
    }
    float* B2;
    {
        k_init_bjorck<<<1, 128, 0, stream>>>(W2, V2a);
        float* cur = V2a; float* nxt = V2b;
        for (int it = 0; it < BJ_ITERS; ++it) {
            k_ata128 <<<8, 256, 0, stream>>>(cur, S);
            k_poly128<<<8, 256, 0, stream>>>(S, Mm);
            k_gemm_k128n128<<<8, 256, 0, stream>>>(cur, Mm, nxt, HID);
            float* t = cur; cur = nxt; nxt = t;
        }
        B2 = cur;
    }

    const int gemm_blocks = (NNODES / 16) * 8 / 8;   // 750 blocks of 8 waves
    const int spmm_blocks = NEDGES / 8;              // 8 waves (edges) per block

    // --- layer 1: relu(norm_adj @ (x @ B1)) ---
    k_gemm_k128n128<<<gemm_blocks, 256, 0, stream>>>(x, B1, XW, NNODES);
    hipMemsetAsync(AX, 0, (size_t)NNODES * HID * sizeof(float), stream);
    k_spmm<<<spmm_blocks, 256, 0, stream>>>(er, ec, dinv, XW, AX, NEDGES);
    k_relu<<<(NNODES * HID + 255) / 256, 256, 0, stream>>>(AX, NNODES * HID);

    // --- layer 2 ---
    k_gemm_k128n128<<<gemm_blocks, 256, 0, stream>>>(AX, B2, HW, NNODES);
    hipMemsetAsync(AH, 0, (size_t)NNODES * HID * sizeof(float), stream);
    k_spmm<<<spmm_blocks, 256, 0, stream>>>(er, ec, dinv, HW, AH, NEDGES);
    k_relu<<<(NNODES * HID + 255) / 256, 256, 0, stream>>>(AH, NNODES * HID);

    // --- head + log_softmax ---
    k_head<<<NNODES, 64, 0, stream>>>(AH, lw, lb, out);
}